// Bilinear_86328842650062
// MI455X (gfx1250) — compile-verified
//
#include <hip/hip_runtime.h>
#include <hip/hip_bf16.h>

typedef __attribute__((ext_vector_type(2))) float v2f;
typedef __attribute__((ext_vector_type(8))) float v8f;

#define S_DIM 4
#define N_DIM 4096
#define D_DIM 64
#define JT 4          // 16-col tiles per wave
#define WAVES_PER_BLOCK 8

// ---------------------------------------------------------------------------
// Kernel 1: Q[s,j,d] = sum_e X0[s,j,e] * W[d,e]   (Q = X0 @ W^T per slice)
// W (64x64 f32, 16KB) cached in LDS with row stride 65 to avoid 64-way bank
// conflicts (address = d*65+e -> bank (d+e)%64, distinct across d).
// ---------------------------------------------------------------------------
__global__ __launch_bounds__(256) void bilinear_q_kernel(
    const float* __restrict__ x0, const float* __restrict__ w,
    float* __restrict__ q) {
  __shared__ float wlds[64 * 65];
  const int t = threadIdx.x;
  for (int i = t; i < 64 * 64; i += 256) {
    wlds[(i >> 6) * 65 + (i & 63)] = w[i];
  }
  __syncthreads();

  const int gid = blockIdx.x * 256 + t;   // over S*N*D = 1,048,576
  const int d = gid & 63;
  const float* __restrict__ xr = x0 + (size_t)(gid >> 6) * D_DIM; // row (s,j)
  const float* __restrict__ wr = wlds + d * 65;
  float acc = 0.f;
#pragma unroll 8
  for (int e = 0; e < D_DIM; ++e) acc = fmaf(xr[e], wr[e], acc);
  q[gid] = acc;
}

// ---------------------------------------------------------------------------
// Kernel 2: out[b,s,i,j] = sum_d X1[s,i,d] * Q[s,j,d] + bias, b = 0,1
// f32 WMMA 16x16x4, K = 64 -> 16 wmma per 16x16 tile.
// Wave layout (per ISA 7.12.2):
//   A 16x4 step kk : lane(lo,hi) = { X1[m0+lo, 4kk+2hi], X1[m0+lo, 4kk+2hi+1] }
//   B 4x16 step kk : lane(lo,hi) = { Q[n0+lo, 4kk+2hi],  Q[n0+lo, 4kk+2hi+1] }
//   D 16x16        : vgpr v, lane(lo,hi) = out row m0+8hi+v, col n0+lo
// ---------------------------------------------------------------------------
__global__ __launch_bounds__(256) void bilinear_wmma_kernel(
    const float* __restrict__ x1, const float* __restrict__ q,
    const float* __restrict__ bias, float* __restrict__ out) {
  const int lane = threadIdx.x & 31;
  const int wave = threadIdx.x >> 5;                 // 0..7
  const int lo = lane & 15;
  const int hi = lane >> 4;

  const int s = blockIdx.z;                          // 0..3
  const int m0 = blockIdx.y * 16;                    // row-tile base
  const int n_base = (blockIdx.x * WAVES_PER_BLOCK + wave) * (JT * 16);

  const float bval = bias[0];
  const float* __restrict__ x1s = x1 + (size_t)s * N_DIM * D_DIM;
  const float* __restrict__ qs = q + (size_t)s * N_DIM * D_DIM;

  // Preload full A fragment for this wave's 16-row strip (16x64 f32).
  v2f a[16];
  {
    const float* __restrict__ ap = x1s + (size_t)(m0 + lo) * D_DIM + 2 * hi;
#pragma unroll
    for (int kk = 0; kk < 16; ++kk) {
      a[kk] = *(const v2f*)(ap + 4 * kk);
    }
  }

  for (int tj = 0; tj < JT; ++tj) {
    const int n0 = n_base + tj * 16;
    const float* __restrict__ bp = qs + (size_t)(n0 + lo) * D_DIM + 2 * hi;

    v8f c = {};
#pragma unroll
    for (int kk = 0; kk < 16; ++kk) {
      v2f bf = *(const v2f*)(bp + 4 * kk);
      // 8 args: (neg_a, A, neg_b, B, c_mod, C, reuse_a, reuse_b)
      c = __builtin_amdgcn_wmma_f32_16x16x4_f32(
          false, a[kk], false, bf, (short)0, c, false, false);
    }

    // Store tile to both batch copies; NT hint: 512MB stream > 192MB L2.
    const int col = n0 + lo;
    const int rowb = m0 + 8 * hi;
    float* __restrict__ o0 =
        out + ((size_t)s * N_DIM + rowb) * N_DIM + col;
    float* __restrict__ o1 = o0 + (size_t)S_DIM * N_DIM * N_DIM;
#pragma unroll
    for (int v = 0; v < 8; ++v) {
      const float val = c[v] + bval;
      __builtin_nontemporal_store(val, o0 + (size_t)v * N_DIM);
      __builtin_nontemporal_store(val, o1 + (size_t)v * N_DIM);
    }
  }
}

extern "C" void kernel_launch(void* const* d_in, const int* in_sizes, int n_in,
                              void* d_out, int out_size, void* d_ws,
                              size_t ws_size, hipStream_t stream) {
  const float* x0 = (const float*)d_in[0];     // (4, 4096, 64)
  const float* x1 = (const float*)d_in[1];     // (4, 4096, 64)
  const float* kern = (const float*)d_in[2];   // (64, 64)
  const float* bias = (const float*)d_in[3];   // scalar
  float* out = (float*)d_out;                  // (2, 4, 4096, 4096)
  float* q = (float*)d_ws;                     // (4, 4096, 64) = 4 MB scratch

  // Stage 1: Q = X0 @ W^T per slice.
  const int q_elems = S_DIM * N_DIM * D_DIM;
  bilinear_q_kernel<<<q_elems / 256, 256, 0, stream>>>(x0, kern, q);

  // Stage 2: out = X1 @ Q^T per slice via f32 WMMA, duplicated to batch=2.
  dim3 grid(N_DIM / (WAVES_PER_BLOCK * JT * 16),  // 8 column super-blocks
            N_DIM / 16,                           // 256 row tiles
            S_DIM);                               // 4 slices
  bilinear_wmma_kernel<<<grid, 256, 0, stream>>>(x1, q, bias, out);
}